// GNN_15281493639464
// MI455X (gfx1250) — compile-verified
//
#include <hip/hip_runtime.h>
#include <hip/hip_bf16.h>

typedef __attribute__((ext_vector_type(2))) float v2f;
typedef __attribute__((ext_vector_type(8))) float v8f;

#define EMB 128
#define HID 32
#define BN_EPS 1e-5f

// ---------------------------------------------------------------------------
// 1) atom_types = argmax(x,1)+1 ; h = atom_emb[atom_types]  (row0 never hit)
//    one wave (32 lanes) per node, first-occurrence argmax over 119 features
// ---------------------------------------------------------------------------
__global__ void GNN_embed(const float* __restrict__ x, const float* __restrict__ emb,
                          float* __restrict__ h, int n) {
    int wid  = (int)((blockIdx.x * (size_t)blockDim.x + threadIdx.x) >> 5);
    int lane = threadIdx.x & 31;
    if (wid >= n) return;
    float bv = -1e30f; int bi = 0x7fffffff;
    const float* xr = x + (size_t)wid * 119;
    for (int j = lane; j < 119; j += 32) {
        float v = xr[j];
        if (v > bv) { bv = v; bi = j; }           // ascending j -> first max in lane
    }
    for (int off = 16; off >= 1; off >>= 1) {
        float ov = __shfl_xor(bv, off, 32);
        int   oi = __shfl_xor(bi, off, 32);
        if (ov > bv || (ov == bv && oi < bi)) { bv = ov; bi = oi; }
    }
    int atom = bi + 1;
    float4 ev = *(const float4*)(emb + (size_t)atom * EMB + lane * 4);
    *(float4*)(h + (size_t)wid * EMB + lane * 4) = ev;
}

// ---------------------------------------------------------------------------
// 2) degree / symmetric norm
// ---------------------------------------------------------------------------
__global__ void GNN_deg_init(float* __restrict__ deg, int n) {
    int i = blockIdx.x * blockDim.x + threadIdx.x;
    if (i < n) deg[i] = 1.0f;                      // self loop contributes 1
}
__global__ void GNN_deg_accum(const int* __restrict__ src, float* __restrict__ deg, int e) {
    int i = blockIdx.x * blockDim.x + threadIdx.x;
    if (i < e) atomicAdd(&deg[src[i]], 1.0f);
}
__global__ void GNN_deg_rsqrt(float* __restrict__ deg, int n) {
    int i = blockIdx.x * blockDim.x + threadIdx.x;
    if (i < n) deg[i] = rsqrtf(deg[i]);            // deg >= 1 always
}
__global__ void GNN_enorm(const int* __restrict__ src, const int* __restrict__ dst,
                          const float* __restrict__ dinv, float* __restrict__ enorm, int e) {
    int i = blockIdx.x * blockDim.x + threadIdx.x;
    if (i < e) enorm[i] = dinv[src[i]] * dinv[dst[i]];
}

// ---------------------------------------------------------------------------
// 3) fused MLP: Y = relu(X @ W1 + b1) @ W2 + b2   (fp32 WMMA 16x16x4)
//    256 threads = 8 waves; one 16-row tile per wave; 128 rows per block
// ---------------------------------------------------------------------------
__global__ __launch_bounds__(256)
void GNN_mlp(const float* __restrict__ X, float* __restrict__ Y,
             const float* __restrict__ W1, const float* __restrict__ B1,
             const float* __restrict__ W2, const float* __restrict__ B2, int n) {
    __shared__ __align__(16) float sW1[EMB * HID];        // [k=128][n=32]
    __shared__ __align__(16) float sW2[HID * EMB];        // [k=32][n=128]
    __shared__ float sB1[HID];
    __shared__ float sB2[EMB];
    __shared__ __align__(16) float sT[8][16][HID + 2];    // per-wave 16x32 intermediate

    for (int i = threadIdx.x; i < EMB * HID; i += 256) { sW1[i] = W1[i]; sW2[i] = W2[i]; }
    if (threadIdx.x < HID) sB1[threadIdx.x] = B1[threadIdx.x];
    if (threadIdx.x < EMB) sB2[threadIdx.x] = B2[threadIdx.x];
    __syncthreads();

    const int wave = threadIdx.x >> 5;
    const int lane = threadIdx.x & 31;
    const int half = lane >> 4;       // 0: K+{0,1},  1: K+{2,3}
    const int l16  = lane & 15;
    const int row0 = blockIdx.x * 128 + wave * 16;

    // ---- stage 1: T = relu(Xtile[16x128] @ W1[128x32] + b1) ----
    int lrow = row0 + l16; if (lrow >= n) lrow = n - 1;   // clamp loads, guard stores
    const float* xr = X + (size_t)lrow * EMB;
    v2f af[32];
    #pragma unroll
    for (int kk = 0; kk < 32; ++kk)
        af[kk] = *(const v2f*)(xr + 4 * kk + 2 * half);

    v8f acc0 = {}; v8f acc1 = {};
    #pragma unroll
    for (int kk = 0; kk < 32; ++kk) {
        int kb = 4 * kk + 2 * half;
        v2f b0, b1v;
        b0.x  = sW1[kb * HID + l16];        b0.y  = sW1[(kb + 1) * HID + l16];
        b1v.x = sW1[kb * HID + 16 + l16];   b1v.y = sW1[(kb + 1) * HID + 16 + l16];
        acc0 = __builtin_amdgcn_wmma_f32_16x16x4_f32(false, af[kk], false, b0,  (short)0, acc0, false, false);
        acc1 = __builtin_amdgcn_wmma_f32_16x16x4_f32(false, af[kk], false, b1v, (short)0, acc1, false, false);
    }
    #pragma unroll
    for (int r = 0; r < 8; ++r) {
        int mm = r + 8 * half;            // C layout: vgpr r -> M=r (lanes 0-15) / M=r+8
        sT[wave][mm][l16]      = fmaxf(acc0[r] + sB1[l16],      0.0f);
        sT[wave][mm][16 + l16] = fmaxf(acc1[r] + sB1[16 + l16], 0.0f);
    }
    // same-wave LDS RAW -> compiler inserts s_wait_dscnt

    // ---- stage 2: Y = T[16x32] @ W2[32x128] + b2 ----
    #pragma unroll
    for (int nt = 0; nt < 8; ++nt) {
        const int n0 = nt * 16;
        v8f acc = {};
        #pragma unroll
        for (int kk = 0; kk < 8; ++kk) {
            int kb = 4 * kk + 2 * half;
            v2f a = *(const v2f*)&sT[wave][l16][kb];
            v2f b;
            b.x = sW2[kb * EMB + n0 + l16];
            b.y = sW2[(kb + 1) * EMB + n0 + l16];
            acc = __builtin_amdgcn_wmma_f32_16x16x4_f32(false, a, false, b, (short)0, acc, false, false);
        }
        #pragma unroll
        for (int r = 0; r < 8; ++r) {
            int orow = row0 + r + 8 * half;
            if (orow < n) Y[(size_t)orow * EMB + n0 + l16] = acc[r] + sB2[n0 + l16];
        }
    }
}

// ---------------------------------------------------------------------------
// 4) aggregation: agg = dinv^2 * m (self loop) then atomic scatter over edges
// ---------------------------------------------------------------------------
__global__ void GNN_self_init(const float* __restrict__ m, const float* __restrict__ dinv,
                              float* __restrict__ agg, int n) {
    size_t tid = blockIdx.x * (size_t)blockDim.x + threadIdx.x;
    int node = (int)(tid >> 5), lane = (int)(tid & 31);
    if (node >= n) return;
    float w = dinv[node]; w *= w;
    float4 v = *(const float4*)(m + (size_t)node * EMB + lane * 4);
    v.x *= w; v.y *= w; v.z *= w; v.w *= w;
    *(float4*)(agg + (size_t)node * EMB + lane * 4) = v;
}
__global__ void GNN_scatter(const float* __restrict__ m, const int* __restrict__ src,
                            const int* __restrict__ dst, const float* __restrict__ enorm,
                            float* __restrict__ agg, int e) {
    size_t tid = blockIdx.x * (size_t)blockDim.x + threadIdx.x;
    int ed = (int)(tid >> 5), lane = (int)(tid & 31);
    if (ed >= e) return;
    int s = src[ed], d = dst[ed];
    float w = enorm[ed];
    float4 v = *(const float4*)(m + (size_t)s * EMB + lane * 4);
    float* o = agg + (size_t)d * EMB + lane * 4;
    atomicAdd(o + 0, w * v.x);
    atomicAdd(o + 1, w * v.y);
    atomicAdd(o + 2, w * v.z);
    atomicAdd(o + 3, w * v.w);
}

// ---------------------------------------------------------------------------
// 5) BatchNorm (training mode, biased var) + optional ReLU
// ---------------------------------------------------------------------------
__global__ void GNN_bn_zero(float* __restrict__ cs, float* __restrict__ cq) {
    int i = threadIdx.x;
    if (i < EMB) { cs[i] = 0.0f; cq[i] = 0.0f; }
}
__global__ void GNN_bn_reduce(const float* __restrict__ hl, float* __restrict__ cs,
                              float* __restrict__ cq, int n) {
    int f = threadIdx.x;                      // 128 threads: one column each
    int r0 = blockIdx.x * 256;
    float s = 0.0f, q = 0.0f;
    for (int r = 0; r < 256; ++r) {
        int row = r0 + r;
        if (row < n) { float v = hl[(size_t)row * EMB + f]; s += v; q += v * v; }
    }
    atomicAdd(&cs[f], s);
    atomicAdd(&cq[f], q);
}
__global__ void GNN_bn_apply(const float* __restrict__ hl, const float* __restrict__ cs,
                             const float* __restrict__ cq, const float* __restrict__ gamma,
                             const float* __restrict__ beta, float* __restrict__ out,
                             int n, int do_relu) {
    size_t tid = blockIdx.x * (size_t)blockDim.x + threadIdx.x;
    if (tid >= (size_t)n * EMB) return;
    int f = (int)(tid & (EMB - 1));
    float invn = 1.0f / (float)n;
    float mu  = cs[f] * invn;
    float var = cq[f] * invn - mu * mu;
    float a = gamma[f] * rsqrtf(var + BN_EPS);
    float b = beta[f] - mu * a;
    float v = a * hl[tid] + b;
    if (do_relu) v = fmaxf(v, 0.0f);
    out[tid] = v;
}

// ---------------------------------------------------------------------------
extern "C" void kernel_launch(void* const* d_in, const int* in_sizes, int n_in,
                              void* d_out, int out_size, void* d_ws, size_t ws_size,
                              hipStream_t stream) {
    const float* x     = (const float*)d_in[0];
    const int*   ei    = (const int*)  d_in[1];
    // d_in[2] = edge_attr (unused by reference output)
    const float* emb   = (const float*)d_in[3];
    const float* W1    = (const float*)d_in[4];
    const float* B1    = (const float*)d_in[5];
    const float* W2    = (const float*)d_in[6];
    const float* B2    = (const float*)d_in[7];
    const float* gamma = (const float*)d_in[8];
    const float* beta  = (const float*)d_in[9];
    float* out = (float*)d_out;

    const int N = in_sizes[0] / 119;
    const int E = in_sizes[1] / 2;
    const int* src = ei;
    const int* dst = ei + E;

    float* ws    = (float*)d_ws;
    float* h     = ws;                         // N*128
    float* m     = h   + (size_t)N * EMB;      // N*128 (also reused as hl)
    float* agg   = m   + (size_t)N * EMB;      // N*128
    float* dinv  = agg + (size_t)N * EMB;      // N
    float* enorm = dinv + N;                   // E
    float* csum  = enorm + E;                  // 128
    float* csq   = csum + EMB;                 // 128

    GNN_embed    <<<(N + 7) / 8,      256, 0, stream>>>(x, emb, h, N);
    GNN_deg_init <<<(N + 255) / 256,  256, 0, stream>>>(dinv, N);
    GNN_deg_accum<<<(E + 255) / 256,  256, 0, stream>>>(src, dinv, E);
    GNN_deg_rsqrt<<<(N + 255) / 256,  256, 0, stream>>>(dinv, N);
    GNN_enorm    <<<(E + 255) / 256,  256, 0, stream>>>(src, dst, dinv, enorm, E);

    const int mlp_blocks   = (N + 127) / 128;
    const int node_waves   = (N + 7) / 8;            // N*32 threads / 256
    const int edge_waves   = (int)(((size_t)E * 32 + 255) / 256);
    const int elem_blocks  = (int)(((size_t)N * EMB + 255) / 256);
    const int bnred_blocks = (N + 255) / 256;

    for (int l = 0; l < 5; ++l) {
        const float* w1 = W1 + (size_t)l * EMB * HID;
        const float* b1 = B1 + (size_t)l * HID;
        const float* w2 = W2 + (size_t)l * HID * EMB;
        const float* b2 = B2 + (size_t)l * EMB;

        GNN_mlp      <<<mlp_blocks,   256, 0, stream>>>(h, m, w1, b1, w2, b2, N);
        GNN_self_init<<<node_waves,   256, 0, stream>>>(m, dinv, agg, N);
        GNN_scatter  <<<edge_waves,   256, 0, stream>>>(m, src, dst, enorm, agg, E);
        GNN_mlp      <<<mlp_blocks,   256, 0, stream>>>(agg, m, w1, b1, w2, b2, N);
        GNN_bn_zero  <<<1,            256, 0, stream>>>(csum, csq);
        GNN_bn_reduce<<<bnred_blocks, 128, 0, stream>>>(m, csum, csq, N);
        float* dbuf = (l == 4) ? out : h;
        GNN_bn_apply <<<elem_blocks,  256, 0, stream>>>(m, csum, csq,
                                                        gamma + (size_t)l * EMB,
                                                        beta  + (size_t)l * EMB,
                                                        dbuf, N, (l < 4) ? 1 : 0);
    }
}